// GNNEncoder_9405978378811
// MI455X (gfx1250) — compile-verified
//
#include <hip/hip_runtime.h>
#include <hip/hip_bf16.h>

typedef __attribute__((ext_vector_type(2))) float v2f;
typedef __attribute__((ext_vector_type(8))) float v8f;

// ---------------- zero fill (float4 wide) ----------------
__global__ __launch_bounds__(256) void zero4_kernel(float4* __restrict__ p, long long n4) {
    long long i = (long long)blockIdx.x * blockDim.x + threadIdx.x;
    if (i < n4) p[i] = make_float4(0.f, 0.f, 0.f, 0.f);
}

// ---------------- degree counts (once; same for both layers) ----------------
__global__ __launch_bounds__(256) void deg_kernel(const int* __restrict__ src,
                                                  const int* __restrict__ dst,
                                                  float* __restrict__ cnt_u,
                                                  float* __restrict__ cnt_m,
                                                  int nE) {
    int e = blockIdx.x * blockDim.x + threadIdx.x;
    if (e >= nE) return;
    atomicAdd(&cnt_u[src[e]], 1.0f);
    atomicAdd(&cnt_m[dst[e]], 1.0f);
}

// ---------------- scatter-add of source rows onto dst accumulator ----------------
// thread = (edge, 16B chunk): 16 threads stream one 256B feature row (coalesced),
// then 4x global_atomic_add_f32 into the destination row.
__global__ __launch_bounds__(256) void scatter_kernel(const float* __restrict__ x,
                                                      const int* __restrict__ sidx,
                                                      const int* __restrict__ didx,
                                                      float* __restrict__ agg,
                                                      long long nWork) {
    long long tid = (long long)blockIdx.x * blockDim.x + threadIdx.x;
    if (tid >= nWork) return;
    long long e = tid >> 4;
    int c = (int)(tid & 15);
    long long s = (long long)sidx[e];
    long long d = (long long)didx[e];
    float4 v = *(const float4*)(x + s * 64 + c * 4);
    float* p = agg + d * 64 + c * 4;
    atomicAdd(p + 0, v.x);
    atomicAdd(p + 1, v.y);
    atomicAdd(p + 2, v.z);
    atomicAdd(p + 3, v.w);
}

// ---------------- fused SAGE GEMM: out = act((agg/cnt)@Wl + b + xdst@Wr) ----------------
// One wave32 per 16-row tile; full 16x64 output tile in 4 v8f accumulators.
// f32 WMMA (16x16x4): A lanes 0-15 hold K=k0,k0+1 ; lanes 16-31 hold K=k0+2,k0+3 ; M=lane&15.
// B mirrors with N=lane&15. Weights are row-major [K,64] == required B layout.
template <bool RELU>
__global__ __launch_bounds__(256) void sage_gemm_kernel(const float* __restrict__ agg,
                                                        const float* __restrict__ cnt,
                                                        const float* __restrict__ xdst,
                                                        const float* __restrict__ Wl,
                                                        const float* __restrict__ Wr,
                                                        const float* __restrict__ bias,
                                                        float* __restrict__ out,
                                                        int nrows) {
    const int lane   = threadIdx.x & 31;
    const int waveId = blockIdx.x * (blockDim.x >> 5) + (threadIdx.x >> 5);
    const int rowBase = waveId << 4;
    if (rowBase >= nrows) return;  // wave-uniform: EXEC stays all-ones for WMMA

    const int m     = lane & 15;   // A: row-in-tile, B: col-in-tile
    const int khalf = lane >> 4;   // 0 -> K=k0..k0+1 ; 1 -> K=k0+2..k0+3
    const int row   = rowBase + m;

    // mean divisor: counts are small integers -> hardware v_rcp_f32 (~1 ulp) is plenty,
    // avoids the full IEEE division expansion.
    const float inv = __builtin_amdgcn_rcpf(fmaxf(cnt[row], 1.0f));
    const float* __restrict__ arow = agg + (size_t)row * 64;
    const float* __restrict__ xrow = xdst + (size_t)row * 64;

    v8f acc[4];
#pragma unroll
    for (int nt = 0; nt < 4; ++nt) acc[nt] = (v8f){0.f, 0.f, 0.f, 0.f, 0.f, 0.f, 0.f, 0.f};

    // (agg * inv) @ Wl
#pragma unroll
    for (int k0 = 0; k0 < 64; k0 += 4) {
        const int ka = k0 + khalf * 2;
        v2f a;
        a.x = arow[ka] * inv;
        a.y = arow[ka + 1] * inv;
#pragma unroll
        for (int nt = 0; nt < 4; ++nt) {
            const int col = nt * 16 + m;
            v2f b;
            b.x = Wl[(size_t)ka * 64 + col];
            b.y = Wl[(size_t)(ka + 1) * 64 + col];
            acc[nt] = __builtin_amdgcn_wmma_f32_16x16x4_f32(
                false, a, false, b, (short)0, acc[nt], false, false);
        }
    }

    // + xdst @ Wr
#pragma unroll
    for (int k0 = 0; k0 < 64; k0 += 4) {
        const int ka = k0 + khalf * 2;
        v2f a;
        a.x = xrow[ka];
        a.y = xrow[ka + 1];
#pragma unroll
        for (int nt = 0; nt < 4; ++nt) {
            const int col = nt * 16 + m;
            v2f b;
            b.x = Wr[(size_t)ka * 64 + col];
            b.y = Wr[(size_t)(ka + 1) * 64 + col];
            acc[nt] = __builtin_amdgcn_wmma_f32_16x16x4_f32(
                false, a, false, b, (short)0, acc[nt], false, false);
        }
    }

    // epilogue: +bias, optional ReLU, store.
    // C/D layout: lanes 0-15: N=lane, M=vgpr ; lanes 16-31: N=lane-16, M=vgpr+8
#pragma unroll
    for (int nt = 0; nt < 4; ++nt) {
        const int col  = nt * 16 + m;
        const float bc = bias[col];
#pragma unroll
        for (int i = 0; i < 8; ++i) {
            const int orow = rowBase + i + khalf * 8;
            float v = acc[nt][i] + bc;
            if (RELU) v = fmaxf(v, 0.0f);
            out[(size_t)orow * 64 + col] = v;
        }
    }
}

static inline int cdiv_i(long long a, long long b) { return (int)((a + b - 1) / b); }

extern "C" void kernel_launch(void* const* d_in, const int* in_sizes, int n_in,
                              void* d_out, int out_size, void* d_ws, size_t ws_size,
                              hipStream_t stream) {
    const float* x_user  = (const float*)d_in[0];
    const float* x_movie = (const float*)d_in[1];
    const int*   e_src   = (const int*)d_in[2];
    const int*   e_dst   = (const int*)d_in[3];
    const float* W1_um_l = (const float*)d_in[4];
    const float* W1_um_r = (const float*)d_in[5];
    const float* W1_mu_l = (const float*)d_in[6];
    const float* W1_mu_r = (const float*)d_in[7];
    const float* W2_um_l = (const float*)d_in[8];
    const float* W2_um_r = (const float*)d_in[9];
    const float* W2_mu_l = (const float*)d_in[10];
    const float* W2_mu_r = (const float*)d_in[11];
    const float* b1_um   = (const float*)d_in[12];
    const float* b1_mu   = (const float*)d_in[13];
    const float* b2_um   = (const float*)d_in[14];
    const float* b2_mu   = (const float*)d_in[15];

    const int N_U = in_sizes[0] / 64;
    const int N_M = in_sizes[1] / 64;
    const int E   = in_sizes[2];

    // workspace layout (floats)
    float* ws    = (float*)d_ws;
    float* cnt_u = ws;                        // N_U
    float* cnt_m = cnt_u + N_U;               // N_M
    float* agg_u = cnt_m + N_M;               // N_U*64
    float* agg_m = agg_u + (size_t)N_U * 64;  // N_M*64
    float* u1    = agg_m + (size_t)N_M * 64;  // N_U*64
    float* m1    = u1 + (size_t)N_U * 64;     // N_M*64

    float* u2 = (float*)d_out;                // [N_U,64]
    float* m2 = u2 + (size_t)N_U * 64;        // [N_M,64]

    const int T = 256;
    const long long scatterWork = (long long)E * 16;
    const int scatterBlocks = cdiv_i(scatterWork, T);

    auto zero = [&](float* p, long long n) {
        zero4_kernel<<<cdiv_i(n / 4, T), T, 0, stream>>>((float4*)p, n / 4);
    };
    auto gemmBlocks = [&](int nrows) { return cdiv_i((nrows + 15) / 16, T / 32); };

    // degrees (shared by both layers)
    zero(cnt_u, (long long)N_U + N_M);
    deg_kernel<<<cdiv_i(E, T), T, 0, stream>>>(e_src, e_dst, cnt_u, cnt_m, E);

    // ---- layer 1 ----
    // m1 = relu(mean_{dst} x_user[src] @ W1_um_l + b1_um + x_movie @ W1_um_r)
    zero(agg_m, (long long)N_M * 64);
    scatter_kernel<<<scatterBlocks, T, 0, stream>>>(x_user, e_src, e_dst, agg_m, scatterWork);
    sage_gemm_kernel<true><<<gemmBlocks(N_M), T, 0, stream>>>(
        agg_m, cnt_m, x_movie, W1_um_l, W1_um_r, b1_um, m1, N_M);

    // u1 = relu(mean_{src} x_movie[dst] @ W1_mu_l + b1_mu + x_user @ W1_mu_r)
    zero(agg_u, (long long)N_U * 64);
    scatter_kernel<<<scatterBlocks, T, 0, stream>>>(x_movie, e_dst, e_src, agg_u, scatterWork);
    sage_gemm_kernel<true><<<gemmBlocks(N_U), T, 0, stream>>>(
        agg_u, cnt_u, x_user, W1_mu_l, W1_mu_r, b1_mu, u1, N_U);

    // ---- layer 2 ----
    // m2 = mean_{dst} u1[src] @ W2_um_l + b2_um + m1 @ W2_um_r
    zero(agg_m, (long long)N_M * 64);
    scatter_kernel<<<scatterBlocks, T, 0, stream>>>(u1, e_src, e_dst, agg_m, scatterWork);
    sage_gemm_kernel<false><<<gemmBlocks(N_M), T, 0, stream>>>(
        agg_m, cnt_m, m1, W2_um_l, W2_um_r, b2_um, m2, N_M);

    // u2 = mean_{src} m1[dst] @ W2_mu_l + b2_mu + u1 @ W2_mu_r
    zero(agg_u, (long long)N_U * 64);
    scatter_kernel<<<scatterBlocks, T, 0, stream>>>(m1, e_dst, e_src, agg_u, scatterWork);
    sage_gemm_kernel<false><<<gemmBlocks(N_U), T, 0, stream>>>(
        agg_u, cnt_u, u1, W2_mu_l, W2_mu_r, b2_mu, u2, N_U);
}